// SpatialTemporalConv_19542101197514
// MI455X (gfx1250) — compile-verified
//
#include <hip/hip_runtime.h>
#include <hip/hip_bf16.h>
#include <math.h>

// Problem constants (match reference)
#define BB 4
#define TT 8192
#define NN 64
#define FF 16
#define NF 1024      // N*F
#define OO 32
#define K3 3

typedef float v2f __attribute__((ext_vector_type(2)));
typedef float v8f __attribute__((ext_vector_type(8)));

// ---------------------------------------------------------------------------
// Kernel 1: one pass over x.
//   rhs[b,n,t]  = sum_f W3[f] * x[b,t,n,f]
//   lhsF[b,n,f] = sum_t W1[t] * x[b,t,n,f]
// grid = B*N blocks, 256 threads; each block owns one (b,n) and strides T.
// ---------------------------------------------------------------------------
__global__ __launch_bounds__(256) void k_reduce(const float* __restrict__ x,
                                                const float* __restrict__ W1,
                                                const float* __restrict__ W3,
                                                float* __restrict__ rhs,
                                                float* __restrict__ lhsF) {
    const int b = blockIdx.x >> 6;
    const int n = blockIdx.x & 63;
    const int tid = threadIdx.x;

    const float* xp = x + (size_t)b * TT * NF + n * FF;
    float* rb = rhs + (size_t)(b * NN + n) * TT;

    float w3[FF];
#pragma unroll
    for (int f = 0; f < FF; ++f) w3[f] = W3[f];

    float acc[FF];
#pragma unroll
    for (int f = 0; f < FF; ++f) acc[f] = 0.f;

    for (int t = tid; t < TT; t += 256) {
        const float* pf = xp + (size_t)t * NF;
        float xv[FF];
#pragma unroll
        for (int f = 0; f < FF; ++f) xv[f] = pf[f];   // 4x b128 loads
        const float w1 = W1[t];
        float d = 0.f;
#pragma unroll
        for (int f = 0; f < FF; ++f) {
            d += xv[f] * w3[f];
            acc[f] += xv[f] * w1;
        }
        rb[t] = d;
    }

    __shared__ float red[256 * FF];   // 16 KB
#pragma unroll
    for (int f = 0; f < FF; ++f) red[tid * FF + f] = acc[f];
    __syncthreads();
    if (tid < FF) {
        float s = 0.f;
        for (int i = 0; i < 256; ++i) s += red[i * FF + tid];
        lhsF[(b * NN + n) * FF + tid] = s;
    }
}

// ---------------------------------------------------------------------------
// Kernel 2a: P[b,f,m] = sum_t W2[f,t] * rhs[b,m,t]
// grid = B*N blocks (one per (b,m)), 256 threads stride T (coalesced).
// ---------------------------------------------------------------------------
__global__ __launch_bounds__(256) void k_pmat(const float* __restrict__ rhs,
                                              const float* __restrict__ W2,
                                              float* __restrict__ P) {
    const int b = blockIdx.x >> 6;
    const int m = blockIdx.x & 63;
    const int tid = threadIdx.x;
    const float* rb = rhs + (size_t)(b * NN + m) * TT;

    float acc[FF];
#pragma unroll
    for (int f = 0; f < FF; ++f) acc[f] = 0.f;

    for (int t = tid; t < TT; t += 256) {
        const float r = rb[t];
#pragma unroll
        for (int f = 0; f < FF; ++f) acc[f] += W2[f * TT + t] * r;
    }

    __shared__ float red[256 * FF];
#pragma unroll
    for (int f = 0; f < FF; ++f) red[tid * FF + f] = acc[f];
    __syncthreads();
    if (tid < FF) {
        float s = 0.f;
        for (int i = 0; i < 256; ++i) s += red[i * FF + tid];
        P[(b * FF + tid) * NN + m] = s;
    }
}

// ---------------------------------------------------------------------------
// Kernel 2b: per batch — product = lhsF @ P, sigmoid(+bs), Spre = Vs @ sig,
// column softmax over n -> S[b,n,k].  grid = B, 256 threads, all in LDS.
// ---------------------------------------------------------------------------
__global__ __launch_bounds__(256) void k_attn(const float* __restrict__ lhsF,
                                              const float* __restrict__ P,
                                              const float* __restrict__ bs,
                                              const float* __restrict__ Vs,
                                              float* __restrict__ S) {
    const int b = blockIdx.x;
    const int tid = threadIdx.x;

    __shared__ float lf[NN * FF];          // 4 KB
    __shared__ float pp[FF * NN];          // 4 KB
    __shared__ float sg[NN * 65];          // sigmoid(product+bs), padded
    __shared__ float sp[NN * 65];          // pre-softmax

    for (int i = tid; i < NN * FF; i += 256) {
        lf[i] = lhsF[b * NN * FF + i];
        pp[i] = P[b * FF * NN + i];
    }
    __syncthreads();

    for (int idx = tid; idx < NN * NN; idx += 256) {
        const int n = idx >> 6, m = idx & 63;
        float s = 0.f;
#pragma unroll
        for (int f = 0; f < FF; ++f) s += lf[n * FF + f] * pp[f * NN + m];
        s += bs[n * NN + m];
        sg[n * 65 + m] = 1.f / (1.f + __expf(-s));
    }
    __syncthreads();

    for (int idx = tid; idx < NN * NN; idx += 256) {
        const int n = idx >> 6, kc = idx & 63;
        float s = 0.f;
        for (int m = 0; m < NN; ++m) s += Vs[n * NN + m] * sg[m * 65 + kc];
        sp[n * 65 + kc] = s;
    }
    __syncthreads();

    if (tid < NN) {                         // softmax over axis=1 (n), one k/thread
        const int kc = tid;
        float mx = -1e30f;
        for (int n = 0; n < NN; ++n) mx = fmaxf(mx, sp[n * 65 + kc]);
        float sum = 0.f;
        for (int n = 0; n < NN; ++n) sum += __expf(sp[n * 65 + kc] - mx);
        const float inv = 1.f / sum;
        for (int n = 0; n < NN; ++n)
            S[(b * NN + n) * NN + kc] = __expf(sp[n * 65 + kc] - mx) * inv;
    }
}

// ---------------------------------------------------------------------------
// Kernel 2c: fold attention + diag(adj) into the conv weight, stored
// QUAD-INTERLEAVED for b64 B-fragment loads:
//   WeffQ[b][k][c>>2][o][c&3] = sum_q S[b,r,q]*adj[q,q]*conv_w[o, q*F+f, k]
// where c = r*F + f.  grid = B*N blocks (one per (b,r)), 256 threads.
// ---------------------------------------------------------------------------
__global__ __launch_bounds__(256) void k_weff(const float* __restrict__ S,
                                              const float* __restrict__ adj,
                                              const float* __restrict__ conv_w,
                                              float* __restrict__ WeffQ) {
    const int b = blockIdx.x >> 6;
    const int r = blockIdx.x & 63;
    const int tid = threadIdx.x;

    __shared__ float sq[NN];
    if (tid < NN) sq[tid] = S[(b * NN + r) * NN + tid] * adj[tid * NN + tid];
    __syncthreads();

    for (int idx = tid; idx < FF * K3 * OO; idx += 256) {
        const int o = idx & 31;
        const int k = (idx >> 5) % K3;
        const int f = idx / (K3 * OO);
        float s = 0.f;
        for (int q = 0; q < NN; ++q)
            s += sq[q] * conv_w[((size_t)o * NF + q * FF + f) * K3 + k];
        const int c = r * FF + f;                  // channel index
        WeffQ[(((size_t)b * K3 + k) * (NF / 4) + (c >> 2)) * (OO * 4)
              + o * 4 + (c & 3)] = s;
    }
}

// ---------------------------------------------------------------------------
// Kernel 3 (hot): out[b,t,o] = sum_k sum_c x[b,t+k-1,c]*Weff[b][k][c][o]
// GEMM M=T, N=32, K=3072 per batch via V_WMMA_F32_16X16X4_F32.
// Block = 64 threads (2 waves); each wave computes 32(t) x 32(o) with FOUR
// v8f accumulators -> per k-step: 2 ds_load_b64 (A) + 2 global_load_b64 (B)
// + 4 WMMAs.  x staged in LDS in 128-channel chunks (66 rows, pitch 132).
// ---------------------------------------------------------------------------
__global__ __launch_bounds__(64) void k_conv(const float* __restrict__ x,
                                             const float* __restrict__ WeffQ,
                                             float* __restrict__ out) {
    const int b = blockIdx.x >> 7;            // 128 t-blocks per batch
    const int tblk = (blockIdx.x & 127) * 64;
    const int tid = threadIdx.x;
    const int wv = tid >> 5;                  // wave 0/1 -> t rows [32wv, 32wv+32)
    const int lane = tid & 31;
    const int lr = lane & 15;
    const int hi = lane >> 4;

    __shared__ __align__(16) float xs[66 * 132];   // 34.8 KB

    v8f acc00 = {0.f,0.f,0.f,0.f,0.f,0.f,0.f,0.f};  // rows+0..15,  o 0..15
    v8f acc01 = {0.f,0.f,0.f,0.f,0.f,0.f,0.f,0.f};  // rows+0..15,  o 16..31
    v8f acc10 = {0.f,0.f,0.f,0.f,0.f,0.f,0.f,0.f};  // rows+16..31, o 0..15
    v8f acc11 = {0.f,0.f,0.f,0.f,0.f,0.f,0.f,0.f};  // rows+16..31, o 16..31

    for (int c0 = 0; c0 < NF; c0 += 128) {
        // --- stage x[tblk-1 .. tblk+64][c0 .. c0+127] into LDS (halo zeroed)
        for (int idx = tid; idx < 66 * 32; idx += 64) {
            const int row = idx >> 5;
            const int cv = idx & 31;
            const int t = tblk - 1 + row;
            float4 v = make_float4(0.f, 0.f, 0.f, 0.f);
            if (t >= 0 && t < TT)
                v = *(const float4*)(x + ((size_t)b * TT + t) * NF + c0 + cv * 4);
            *(float4*)(xs + row * 132 + cv * 4) = v;
        }
        // hint the next chunk into cache while we compute this one
        if (c0 + 128 < NF && tid < 64) {
            const int t = tblk - 1 + tid;
            if (t >= 0 && t < TT)
                __builtin_prefetch(x + ((size_t)b * TT + t) * NF + c0 + 128, 0, 3);
        }
        __syncthreads();

        // --- 3 taps x 32 k-steps, 4 WMMAs per step
#pragma unroll
        for (int k = 0; k < K3; ++k) {
            const float* xrow0 = xs + (32 * wv + lr + k) * 132 + 2 * hi;
            const float* xrow1 = xrow0 + 16 * 132;
            // quad-interleaved weights: lane's two K-values are contiguous
            const float* wpk = WeffQ
                + (((size_t)b * K3 + k) * (NF / 4) + (c0 >> 2)) * (OO * 4)
                + lr * 4 + 2 * hi;
#pragma unroll 4
            for (int cs = 0; cs < 32; ++cs) {
                v2f a0 = *(const v2f*)(xrow0 + cs * 4);       // ds_load_b64
                v2f a1 = *(const v2f*)(xrow1 + cs * 4);       // ds_load_b64
                v2f b0 = *(const v2f*)(wpk + (size_t)cs * (OO * 4));        // b64
                v2f b1 = *(const v2f*)(wpk + (size_t)cs * (OO * 4) + 64);   // b64
                acc00 = __builtin_amdgcn_wmma_f32_16x16x4_f32(
                            false, a0, false, b0, (short)0, acc00, false, false);
                acc01 = __builtin_amdgcn_wmma_f32_16x16x4_f32(
                            false, a0, false, b1, (short)0, acc01, false, false);
                acc10 = __builtin_amdgcn_wmma_f32_16x16x4_f32(
                            false, a1, false, b0, (short)0, acc10, false, false);
                acc11 = __builtin_amdgcn_wmma_f32_16x16x4_f32(
                            false, a1, false, b1, (short)0, acc11, false, false);
            }
        }
        __syncthreads();
    }

    // --- C/D layout: VGPR j -> M = j + 8*hi, N = lr (per 16x16 tile)
    const int trow = tblk + 32 * wv + 8 * hi;
    float* ob = out + ((size_t)b * TT + trow) * OO + lr;
#pragma unroll
    for (int j = 0; j < 8; ++j) {
        ob[(size_t)j * OO]             = acc00[j];
        ob[(size_t)j * OO + 16]        = acc01[j];
        ob[(size_t)(16 + j) * OO]      = acc10[j];
        ob[(size_t)(16 + j) * OO + 16] = acc11[j];
    }
}

// ---------------------------------------------------------------------------
extern "C" void kernel_launch(void* const* d_in, const int* in_sizes, int n_in,
                              void* d_out, int out_size, void* d_ws, size_t ws_size,
                              hipStream_t stream) {
    const float* x      = (const float*)d_in[0];
    const float* adj    = (const float*)d_in[1];
    const float* W1     = (const float*)d_in[2];
    const float* W2     = (const float*)d_in[3];
    const float* W3     = (const float*)d_in[4];
    const float* bs     = (const float*)d_in[5];
    const float* Vs     = (const float*)d_in[6];
    const float* conv_w = (const float*)d_in[7];
    float* out = (float*)d_out;

    // Workspace layout (floats): rhs | lhsF | P | S | WeffQ  (~10.1 MB)
    float* rhs   = (float*)d_ws;
    float* lhsF  = rhs  + (size_t)BB * NN * TT;
    float* P     = lhsF + (size_t)BB * NN * FF;
    float* S     = P    + (size_t)BB * FF * NN;
    float* WeffQ = S    + (size_t)BB * NN * NN;

    k_reduce<<<BB * NN, 256, 0, stream>>>(x, W1, W3, rhs, lhsF);
    k_pmat  <<<BB * NN, 256, 0, stream>>>(rhs, W2, P);
    k_attn  <<<BB,      256, 0, stream>>>(lhsF, P, bs, Vs, S);
    k_weff  <<<BB * NN, 256, 0, stream>>>(S, adj, conv_w, WeffQ);
    k_conv  <<<BB * (TT / 64), 64, 0, stream>>>(x, WeffQ, out);
}